// Treedeliver_spatial_12541304504879
// MI455X (gfx1250) — compile-verified
//
#include <hip/hip_runtime.h>

// Treedeliver_spatial: B=8, T=20, N=512, IN_DIMS=2, EMBED=16 (all f32).
// Memory-bound (AI ~2 FLOP/byte): adjacency = 167.8 MB, read twice; fits L2.
// Pass 0: P = prelu(A @ G)   (WMMA f32 16x16x4, B panel zero-padded to 16 cols)
// Pass 1: U = A @ P + diag(A)*(prelu(G)-P)   (A re-read hits 192 MB L2)
// Pass 2: out = prelu(U @ W^T)   (elementwise, coalesced 64B stores)

typedef __attribute__((ext_vector_type(2))) float v2f;
typedef __attribute__((ext_vector_type(8))) float v8f;

#define NN 512           // nodes per slice
#define IND 2            // input dims
#define EMB 16           // embedding dims

__device__ __forceinline__ float prelu_f(float x, float a) {
    return x >= 0.0f ? x : a * x;
}

// One wave computes one 16-row block of (A_slice @ B_slice). The [N,2] B panel
// is staged TRANSPOSED ([2][N]) in LDS so each lane's two K-values per WMMA
// step are one contiguous ds_load_b64; padding columns 2..15 are zeroed with
// branch-free selects (no EXEC churn in the inner loop).
// PASS 0: Obuf(=Pt) gets prelu(D), stored transposed [2][N] per slice.
// PASS 1: Obuf(=U)  gets D + diag(A)*(prelu(G)-P), stored [node][2].
template <int PASS>
__global__ __launch_bounds__(256) void tree_pass_kernel(
    const float* __restrict__ A,      // [slices, N, N]
    const float* __restrict__ G,      // [slices, N, 2]
    const float* __restrict__ Pt,     // [slices, 2, N] transposed P (pass 1 in)
    float* __restrict__ Obuf,         // pass0: Pt, pass1: Ubuf
    const float* __restrict__ alpha_p)
{
    __shared__ float Bt[IND * NN];    // B panel, transposed: Bt[c*N + k]

    const float alpha = alpha_p[0];
    const int wg   = blockIdx.x;      // 4 WGs per slice
    const int s    = wg >> 2;
    const int tid  = threadIdx.x;
    const int wave = tid >> 5;        // 0..7
    const int lane = tid & 31;
    const int half = lane >> 4;       // 0: lanes 0-15, 1: lanes 16-31
    const int nm   = lane & 15;       // A: row-in-block; B/D: column
    const int row0 = ((((wg & 3) << 3) + wave) << 4);

    const float* As = A + (size_t)s * NN * NN;

    // Stage B panel into LDS (uniform trip counts -> no divergence).
    if constexpr (PASS == 0) {
        const float* Gs = G + (size_t)s * NN * IND;
        for (int i = tid; i < NN; i += 256) {        // transpose on the fly
            v2f g = *(const v2f*)(Gs + i * IND);
            Bt[i]      = g.x;
            Bt[NN + i] = g.y;
        }
    } else {
        const float* Ps = Pt + (size_t)s * NN * IND; // already transposed
        for (int i = tid; i < IND * NN; i += 256)
            Bt[i] = Ps[i];
    }
    __syncthreads();

    const bool valid = nm < IND;                     // loop-invariant mask
    const float* brow = &Bt[(nm & 1) * NN + 2 * half];   // clamped lane addr
    // A operand: row = row0+nm; V0 = K{k0 | k0+2}, V1 = K{k0+1 | k0+3}.
    const float* arow = As + (size_t)(row0 + nm) * NN + 2 * half;

    v8f acc = {};
    #pragma unroll 4
    for (int k0 = 0; k0 < NN; k0 += 4) {
        v2f a  = *(const v2f*)(arow + k0);   // global_load_b64 (A stream)
        v2f bv = *(const v2f*)(brow + k0);   // ds_load_b64 (broadcast)
        v2f b  = { valid ? bv.x : 0.0f, valid ? bv.y : 0.0f };
        acc = __builtin_amdgcn_wmma_f32_16x16x4_f32(
            false, a, false, b, (short)0, acc, false, false);
    }

    // D layout: lane = column, VGPR v -> row v + 8*half. Only cols 0,1 valid.
    if (valid) {
        const int rbase = row0 + half * 8;
        #pragma unroll
        for (int v = 0; v < 8; ++v) {
            const int r = rbase + v;
            if constexpr (PASS == 0) {
                // store transposed: Pt[s][nm][r]
                Obuf[(size_t)s * NN * IND + (size_t)nm * NN + r] =
                    prelu_f(acc[v], alpha);
            } else {
                const float g  = G[(size_t)s * NN * IND + (size_t)r * IND + nm];
                const float p  = Bt[nm * NN + r];          // P from LDS
                const float dg = As[(size_t)r * NN + r];   // A[r,r] (L2 hit)
                Obuf[(size_t)s * NN * IND + (size_t)r * IND + nm] =
                    acc[v] + dg * (prelu_f(g, alpha) - p);
            }
        }
    }
}

// out[node, e] = prelu(u0*W[e,0] + u1*W[e,1]); one node per thread, 64B store.
__global__ __launch_bounds__(256) void tree_embed_kernel(
    const float* __restrict__ Ubuf,   // [total_nodes, 2]
    const float* __restrict__ W,      // [16, 2]
    const float* __restrict__ alpha_p,
    float* __restrict__ out,          // [total_nodes, 16]
    int total_nodes)
{
    const float alpha = alpha_p[0];
    const int node = blockIdx.x * blockDim.x + threadIdx.x;
    if (node >= total_nodes) return;
    const float u0 = Ubuf[(size_t)node * IND + 0];
    const float u1 = Ubuf[(size_t)node * IND + 1];
    float o[EMB];
    #pragma unroll
    for (int e = 0; e < EMB; ++e)
        o[e] = prelu_f(u0 * W[e * IND + 0] + u1 * W[e * IND + 1], alpha);
    float4* dst = (float4*)(out + (size_t)node * EMB);
    #pragma unroll
    for (int q = 0; q < 4; ++q)
        dst[q] = make_float4(o[4 * q], o[4 * q + 1], o[4 * q + 2], o[4 * q + 3]);
}

extern "C" void kernel_launch(void* const* d_in, const int* in_sizes, int n_in,
                              void* d_out, int out_size, void* d_ws, size_t ws_size,
                              hipStream_t stream) {
    (void)n_in; (void)out_size; (void)ws_size;
    const float* graph = (const float*)d_in[0];   // [B,T,N,2]
    const float* adj   = (const float*)d_in[1];   // [B,T,N,N]
    const float* W     = (const float*)d_in[2];   // [16,2]
    const float* alpha = (const float*)d_in[3];   // scalar
    float* out = (float*)d_out;

    const int slices = in_sizes[0] / (NN * IND);  // B*T = 160
    const int total_nodes = slices * NN;

    float* Ptws = (float*)d_ws;                   // [slices, 2, N]   640 KB
    float* Ubuf = Ptws + (size_t)total_nodes * IND; // [nodes, 2]     640 KB

    dim3 block(256);
    dim3 grid(slices * 4);                        // 8 waves/WG, 1 row-block/wave
    tree_pass_kernel<0><<<grid, block, 0, stream>>>(adj, graph, Ptws, Ptws, alpha);
    tree_pass_kernel<1><<<grid, block, 0, stream>>>(adj, graph, Ptws, Ubuf, alpha);
    tree_embed_kernel<<<(total_nodes + 255) / 256, block, 0, stream>>>(
        Ubuf, W, alpha, out, total_nodes);
}